// MPNNs_24266565222959
// MI455X (gfx1250) — compile-verified
//
#include <hip/hip_runtime.h>
#include <hip/hip_bf16.h>

#define N_NODES 50000
#define N_EDGES 800000
#define HID 128
#define LAYERS 3
#define LN_EPS 1e-5f

typedef __attribute__((ext_vector_type(2))) float v2f;
typedef __attribute__((ext_vector_type(8))) float v8f;
typedef __attribute__((ext_vector_type(4))) unsigned int v4u;
typedef __attribute__((ext_vector_type(4))) int v4i;
typedef __attribute__((ext_vector_type(8))) int v8i;

// ---------------------------------------------------------------------------
// Degree / norm kernels
// ---------------------------------------------------------------------------
__global__ void deg_init_kernel(float* __restrict__ deg, int n) {
    int i = blockIdx.x * blockDim.x + threadIdx.x;
    if (i < n) deg[i] = 1.0f;  // self loop contributes 1
}

__global__ void deg_accum_kernel(const long long* __restrict__ ei,
                                 float* __restrict__ deg, int e) {
    int i = blockIdx.x * blockDim.x + threadIdx.x;
    if (i < e) {
        int d = (int)ei[(size_t)e + i];  // dst row of edge_index
        atomicAdd(&deg[d], 1.0f);
    }
}

__global__ void deg_finalize_kernel(float* __restrict__ deg, int n) {
    int i = blockIdx.x * blockDim.x + threadIdx.x;
    if (i < n) deg[i] = rsqrtf(deg[i]);  // deg >= 1 always
}

// ---------------------------------------------------------------------------
// WMMA fp32 GEMM: OUT[nrows x 128] = X[nrows x 128] @ W[128 x 128] (+ bias)
// Block = 256 threads = 8 waves; block covers 128 rows; wave w owns the
// 16-row tile [blk*128 + 16w, +16) and loops over 8 column tiles, holding
// all 8 f32 accumulator tiles (64 VGPRs) so each A-fragment is loaded once
// per K-step. W (64 KB) is staged into LDS by the Tensor Data Mover:
// one 128x128 f32 tile descriptor, issued by wave 0, TENSORcnt-tracked.
// ---------------------------------------------------------------------------
__global__ __launch_bounds__(256) void gemm128_kernel(
    const float* __restrict__ X, const float* __restrict__ W,
    const float* __restrict__ bias, float* __restrict__ OUT, int nrows) {
    __shared__ float Ws[HID * HID];  // 64 KB

    const int tid = threadIdx.x;

    if ((tid >> 5) == 0) {
        // Build Tensor DMA Descriptor (D#) per CDNA5 ISA.
        // Flat shared-aperture address: low 32 bits == LDS byte offset.
        unsigned int lds_off = (unsigned int)(size_t)(&Ws[0]);
        unsigned long long ga = (unsigned long long)(size_t)W;
        // Group0: count=1 | lds_addr | global_addr[56:0] | type=2
        v4u g0;
        g0.x = 1u;                                   // count=1, user mode
        g0.y = lds_off;                              // lds_addr (bytes)
        g0.z = (unsigned int)(ga & 0xffffffffu);     // global_addr[31:0]
        g0.w = (unsigned int)((ga >> 32) & 0x1ffffffu) | (2u << 30);  // +type=2
        // Group1: data_size=4B(2), tensor_dim0=128, tensor_dim1=128,
        //         tile_dim0=128, tile_dim1=128, tile_dim2=0, dim0_stride=128
        v8i g1;
        g1[0] = 0x20000;            // workgroup_mask=0 | data_size=2<<16
        g1[1] = (int)(128u << 16);  // tensor_dim0[15:0] << 16
        g1[2] = (int)(128u << 16);  // tensor_dim0 hi=0 | tensor_dim1[15:0]<<16
        g1[3] = (int)(128u << 16);  // tensor_dim1 hi=0 | tile_dim0=128
        g1[4] = 128;                // tile_dim1=128 | tile_dim2=0
        g1[5] = 128;                // tensor_dim0_stride[31:0] = 128
        g1[6] = 0;                  // stride0 hi | tensor_dim1_stride lo
        g1[7] = 0;                  // tensor_dim1_stride hi
        v4i gz4 = (v4i){0, 0, 0, 0};            // groups 2/3 unused (2D)
        v8i gz8 = (v8i){0, 0, 0, 0, 0, 0, 0, 0};
        __builtin_amdgcn_tensor_load_to_lds(g0, g1, gz4, gz4, gz8, 0);
        __builtin_amdgcn_s_wait_tensorcnt(0);
    }
    __syncthreads();

    const int wave  = tid >> 5;
    const int lane  = tid & 31;
    const int lhalf = lane >> 4;   // 0: lanes 0-15, 1: lanes 16-31
    const int l16   = lane & 15;

    const int row0 = blockIdx.x * 128 + wave * 16;

    v8f acc[8];
    #pragma unroll
    for (int t = 0; t < 8; ++t) acc[t] = (v8f){0, 0, 0, 0, 0, 0, 0, 0};

    // A-fragment row for this lane (clamped so every lane stays in-bounds;
    // junk rows are computed but never stored — keeps EXEC all-ones at WMMA).
    int arow = row0 + l16;
    if (arow >= nrows) arow = nrows - 1;
    const float* __restrict__ xrow = X + (size_t)arow * HID;

    for (int k0 = 0; k0 < HID; k0 += 4) {
        // A 16x4 layout: lane<16 -> K={k0,k0+1}; lane>=16 -> K={k0+2,k0+3}
        v2f a;
        a.x = xrow[k0 + 2 * lhalf + 0];
        a.y = xrow[k0 + 2 * lhalf + 1];
        #pragma unroll
        for (int t = 0; t < 8; ++t) {
            // B 4x16 layout (mirror of C): vgpr0 rows {0,2}, vgpr1 rows {1,3}
            const int ncol = t * 16 + l16;
            v2f b;
            b.x = Ws[(k0 + 2 * lhalf + 0) * HID + ncol];
            b.y = Ws[(k0 + 2 * lhalf + 1) * HID + ncol];
            acc[t] = __builtin_amdgcn_wmma_f32_16x16x4_f32(
                false, a, false, b, (short)0, acc[t], false, false);
        }
    }

    // C/D layout: VGPR r holds row r (lanes 0-15) and row r+8 (lanes 16-31),
    // column = lane & 15.
    #pragma unroll
    for (int t = 0; t < 8; ++t) {
        const int col = t * 16 + l16;
        const float bcol = bias ? bias[col] : 0.0f;
        #pragma unroll
        for (int r = 0; r < 8; ++r) {
            int grow = row0 + r + 8 * lhalf;
            if (grow < nrows)
                OUT[(size_t)grow * HID + col] = acc[t][r] + bcol;
        }
    }
}

// ---------------------------------------------------------------------------
// agg[n][f] = conv_b[f] + dinv[n]^2 * h[n][f]    (self-loop term + bias)
// ---------------------------------------------------------------------------
__global__ void agg_init_kernel(const float* __restrict__ h,
                                const float* __restrict__ dinv,
                                const float* __restrict__ convb,
                                float* __restrict__ agg, int n) {
    int idx = blockIdx.x * blockDim.x + threadIdx.x;
    if (idx < n * HID) {
        int row = idx >> 7;
        int f = idx & (HID - 1);
        float di = dinv[row];
        agg[idx] = convb[f] + di * di * h[idx];
    }
}

// ---------------------------------------------------------------------------
// Edge scatter: one wave per edge; lane owns 4 consecutive features.
// agg[dst] += dinv[src]*dinv[dst] * h[src]
// ---------------------------------------------------------------------------
__global__ __launch_bounds__(256) void scatter_kernel(
    const long long* __restrict__ ei, const float* __restrict__ dinv,
    const float* __restrict__ h, float* __restrict__ agg, int e) {
    int edge = blockIdx.x * (blockDim.x >> 5) + (threadIdx.x >> 5);
    int lane = threadIdx.x & 31;
    if (edge >= e) return;
    int s = (int)ei[edge];
    int d = (int)ei[(size_t)e + edge];
    float w = dinv[s] * dinv[d];
    const float4* __restrict__ hs = (const float4*)(h + (size_t)s * HID);
    float* __restrict__ ad = agg + (size_t)d * HID;
    float4 v = hs[lane];
    atomicAdd(&ad[lane * 4 + 0], v.x * w);
    atomicAdd(&ad[lane * 4 + 1], v.y * w);
    atomicAdd(&ad[lane * 4 + 2], v.z * w);
    atomicAdd(&ad[lane * 4 + 3], v.w * w);
}

// ---------------------------------------------------------------------------
// Fused epilogue: v = agg + tmp; LayerNorm; ReLU; xcur = v; out (JK sum).
// One wave per row (wave32), lane owns 4 features; __shfl_xor reductions.
// ---------------------------------------------------------------------------
__global__ __launch_bounds__(256) void epilogue_kernel(
    const float* __restrict__ agg, const float* __restrict__ tmp,
    const float* __restrict__ g, const float* __restrict__ b,
    float* __restrict__ xcur, float* __restrict__ out, int accumulate,
    int nrows) {
    int row = blockIdx.x * (blockDim.x >> 5) + (threadIdx.x >> 5);
    int lane = threadIdx.x & 31;
    if (row >= nrows) return;

    const float4 a4 = ((const float4*)(agg + (size_t)row * HID))[lane];
    const float4 t4 = ((const float4*)(tmp + (size_t)row * HID))[lane];
    float v0 = a4.x + t4.x, v1 = a4.y + t4.y, v2 = a4.z + t4.z, v3 = a4.w + t4.w;

    float s = v0 + v1 + v2 + v3;
    #pragma unroll
    for (int o = 16; o > 0; o >>= 1) s += __shfl_xor(s, o, 32);
    float mu = s * (1.0f / HID);

    float d0 = v0 - mu, d1 = v1 - mu, d2 = v2 - mu, d3 = v3 - mu;
    float ss = d0 * d0 + d1 * d1 + d2 * d2 + d3 * d3;
    #pragma unroll
    for (int o = 16; o > 0; o >>= 1) ss += __shfl_xor(ss, o, 32);
    float inv = rsqrtf(ss * (1.0f / HID) + LN_EPS);

    const float4 g4 = ((const float4*)g)[lane];
    const float4 b4 = ((const float4*)b)[lane];
    float4 r;
    r.x = fmaxf(d0 * inv * g4.x + b4.x, 0.0f);
    r.y = fmaxf(d1 * inv * g4.y + b4.y, 0.0f);
    r.z = fmaxf(d2 * inv * g4.z + b4.z, 0.0f);
    r.w = fmaxf(d3 * inv * g4.w + b4.w, 0.0f);

    ((float4*)(xcur + (size_t)row * HID))[lane] = r;

    float4* op = (float4*)(out + (size_t)row * HID) + lane;
    if (accumulate) {
        float4 o4 = *op;
        o4.x += r.x; o4.y += r.y; o4.z += r.z; o4.w += r.w;
        *op = o4;
    } else {
        *op = r;
    }
}

// ---------------------------------------------------------------------------
extern "C" void kernel_launch(void* const* d_in, const int* in_sizes, int n_in,
                              void* d_out, int out_size, void* d_ws,
                              size_t ws_size, hipStream_t stream) {
    (void)in_sizes; (void)n_in; (void)out_size; (void)ws_size;
    const float*      x        = (const float*)d_in[0];
    const long long*  ei       = (const long long*)d_in[1];  // int64 [2,E]
    const float*      lin_in_w = (const float*)d_in[2];
    const float*      lin_in_b = (const float*)d_in[3];
    const float*      conv_w   = (const float*)d_in[4];  // [L,128,128]
    const float*      conv_b   = (const float*)d_in[5];  // [L,128]
    const float*      lin_w    = (const float*)d_in[6];  // [L,128,128]
    const float*      lin_b    = (const float*)d_in[7];  // [L,128]
    const float*      ln_g     = (const float*)d_in[8];  // [L,128]
    const float*      ln_b     = (const float*)d_in[9];  // [L,128]
    float*            out      = (float*)d_out;

    // Workspace layout (floats): dinv | xcur | h | agg
    float* ws = (float*)d_ws;
    const size_t NA = (size_t)N_NODES * HID;
    float* dinv = ws;
    float* xcur = ws + 50176;           // 50000 rounded up to 128*392
    float* h    = xcur + NA;
    float* agg  = h + NA;

    const int nblk256 = (N_NODES + 255) / 256;
    const int gemm_blks = (N_NODES + 127) / 128;

    // 1. degree / dinv
    deg_init_kernel<<<nblk256, 256, 0, stream>>>(dinv, N_NODES);
    deg_accum_kernel<<<(N_EDGES + 255) / 256, 256, 0, stream>>>(ei, dinv, N_EDGES);
    deg_finalize_kernel<<<nblk256, 256, 0, stream>>>(dinv, N_NODES);

    // 2. pre-linear: xcur = x @ lin_in_w + lin_in_b
    gemm128_kernel<<<gemm_blks, 256, 0, stream>>>(x, lin_in_w, lin_in_b, xcur,
                                                  N_NODES);

    for (int i = 0; i < LAYERS; ++i) {
        const float* cw = conv_w + (size_t)i * HID * HID;
        const float* cb = conv_b + (size_t)i * HID;
        const float* lw = lin_w + (size_t)i * HID * HID;
        const float* lb = lin_b + (size_t)i * HID;
        const float* lg = ln_g + (size_t)i * HID;
        const float* lbb = ln_b + (size_t)i * HID;

        // h = xcur @ conv_w[i]
        gemm128_kernel<<<gemm_blks, 256, 0, stream>>>(xcur, cw, nullptr, h,
                                                      N_NODES);
        // agg = conv_b + dinv^2 * h (self loops)
        agg_init_kernel<<<(int)((NA + 255) / 256), 256, 0, stream>>>(
            h, dinv, cb, agg, N_NODES);
        // agg += scatter over edges
        scatter_kernel<<<(N_EDGES + 7) / 8, 256, 0, stream>>>(ei, dinv, h, agg,
                                                              N_EDGES);
        // tmp (reuse h) = xcur @ lin_w[i] + lin_b[i]
        gemm128_kernel<<<gemm_blks, 256, 0, stream>>>(xcur, lw, lb, h, N_NODES);
        // LayerNorm + ReLU + JK accumulate
        epilogue_kernel<<<(N_NODES + 7) / 8, 256, 0, stream>>>(
            agg, h, lg, lbb, xcur, out, i > 0 ? 1 : 0, N_NODES);
    }
}